// AGAT_50259707298275
// MI455X (gfx1250) — compile-verified
//
#include <hip/hip_runtime.h>
#include <math.h>

// T=3 types, D=64 feats, L=2 layers (structural constants of the reference)
#define TT 3
#define DD 64

typedef __attribute__((ext_vector_type(2))) float v2f;
typedef __attribute__((ext_vector_type(8))) float v8f;

// ---------------------------------------------------------------------------
// Fill kernels
// ---------------------------------------------------------------------------
__global__ void k_fill(float* __restrict__ p, float v, size_t n) {
  size_t g = (size_t)blockIdx.x * blockDim.x + threadIdx.x;
  if (g < n) p[g] = v;
}
__global__ void k_fill_i32(int* __restrict__ p, int v, size_t n) {
  size_t g = (size_t)blockIdx.x * blockDim.x + threadIdx.x;
  if (g < n) p[g] = v;
}

// ---------------------------------------------------------------------------
// CSR construction (once per launch; edge_index is layer-invariant)
// ---------------------------------------------------------------------------
__global__ void k_count_deg(const int* __restrict__ row, int* __restrict__ deg,
                            int En) {
  int e = blockIdx.x * blockDim.x + threadIdx.x;
  if (e >= En) return;
  atomicAdd(&deg[row[e]], 1);
}

#define SCAN_THREADS 1024
__global__ __launch_bounds__(SCAN_THREADS) void k_scan_rowptr(
    const int* __restrict__ deg, int* __restrict__ rowptr, int Nn) {
  __shared__ int psum[SCAN_THREADS];
  int tid = threadIdx.x;
  int chunk = (Nn + SCAN_THREADS - 1) / SCAN_THREADS;
  int begin = tid * chunk;
  int end = begin + chunk;
  if (end > Nn) end = Nn;
  if (begin > Nn) begin = Nn;
  int s = 0;
  for (int i = begin; i < end; ++i) s += deg[i];
  psum[tid] = s;
  __syncthreads();
  for (int offm = 1; offm < SCAN_THREADS; offm <<= 1) {
    int v = (tid >= offm) ? psum[tid - offm] : 0;
    __syncthreads();
    psum[tid] += v;
    __syncthreads();
  }
  int run = (tid == 0) ? 0 : psum[tid - 1];
  for (int i = begin; i < end; ++i) {
    rowptr[i] = run;
    run += deg[i];
  }
  if (tid == SCAN_THREADS - 1) rowptr[Nn] = run;
}

__global__ void k_bucket_edges(const int* __restrict__ row,
                               const int* __restrict__ rowptr,
                               int* __restrict__ cursor,
                               int* __restrict__ eord, int En) {
  int e = blockIdx.x * blockDim.x + threadIdx.x;
  if (e >= En) return;
  int r = row[e];
  int p = atomicAdd(&cursor[r], 1);
  eord[rowptr[r] + p] = e;
}

// ---------------------------------------------------------------------------
// Small per-layer parameter kernel (single block):
//   efc      = relu?(ef_in)                       [T,D]  (kept in LDS)
//   ef_new   = efc @ wr                           [T,D]
//   sig_ef   = sigmoid(ef_new)                    [T,D]
//   rgtab[t,tp] = sum_d efc[tp,d] * tg[t,d]       [T,T]
// ---------------------------------------------------------------------------
__global__ void k_small_params(const float* __restrict__ ef_in, int relu_in,
                               const float* __restrict__ tg,
                               const float* __restrict__ wr,
                               float* __restrict__ ef_new,
                               float* __restrict__ sig_ef,
                               float* __restrict__ rgtab) {
  __shared__ float efs[TT * DD];
  int tid = threadIdx.x;  // 256 threads
  if (tid < TT * DD) {
    float v = ef_in[tid];
    if (relu_in) v = fmaxf(v, 0.0f);
    efs[tid] = v;
  }
  __syncthreads();
  if (tid < TT * DD) {
    int t = tid / DD, d = tid % DD;
    float acc = 0.0f;
    for (int k = 0; k < DD; ++k) acc += efs[t * DD + k] * wr[k * DD + d];
    ef_new[tid] = acc;
    sig_ef[tid] = 1.0f / (1.0f + expf(-acc));
  } else if (tid < TT * DD + TT * TT) {
    int i = tid - TT * DD;
    int t = i / TT, tp = i % TT;
    float acc = 0.0f;
    for (int d = 0; d < DD; ++d) acc += efs[tp * DD + d] * tg[t * DD + d];
    rgtab[i] = acc;
  }
}

// ---------------------------------------------------------------------------
// Per-node attention projections: ri[t,n] = h[t,n,:].thi[t], rj likewise.
// hstride_t == 0 means h is broadcast x (layer 0).
// ---------------------------------------------------------------------------
__global__ void k_node_proj(const float* __restrict__ H, size_t hstride_t,
                            const float* __restrict__ thi,
                            const float* __restrict__ thj,
                            float* __restrict__ ri, float* __restrict__ rj,
                            int Nn) {
  int n = blockIdx.x * blockDim.x + threadIdx.x;
  int t = blockIdx.y;
  if (n >= Nn) return;
  const float* hr = H + (size_t)t * hstride_t + (size_t)n * DD;
  const float* a = thi + t * DD;
  const float* b = thj + t * DD;
  float si = 0.0f, sj = 0.0f;
  for (int d = 0; d < DD; ++d) {
    float hv = hr[d];
    si += hv * a[d];
    sj += hv * b[d];
  }
  ri[(size_t)t * Nn + n] = si;
  rj[(size_t)t * Nn + n] = sj;
}

// ---------------------------------------------------------------------------
// WMMA f32 GEMM: XW[M,64] = H[M,64] @ W[64,64]  (row-major, full f32)
// One wave per 16-row tile; 4 waves/block; W staged in LDS (16 KB, shared).
// Uses V_WMMA_F32_16X16X4_F32 (exact f32, no precision loss vs reference).
// A 16x4 layout : lanes 0-15 M=0..15 K={0,1}; lanes 16-31 M=0..15 K={2,3}
// B 4x16 layout : lanes 0-15 N=0..15 K={0,1}; lanes 16-31 N=0..15 K={2,3}
// C/D 16x16     : VGPR r -> lanes 0-15 (M=r), lanes 16-31 (M=r+8)
// ---------------------------------------------------------------------------
__global__ __launch_bounds__(128) void k_gemm_we(const float* __restrict__ H,
                                                 const float* __restrict__ W,
                                                 float* __restrict__ XW,
                                                 int mtiles) {
  __shared__ float sW[DD * DD];
  for (int i = threadIdx.x; i < (DD * DD) / 4; i += 128)
    ((float4*)sW)[i] = ((const float4*)W)[i];
  __syncthreads();

  int wave = threadIdx.x >> 5;
  int lane = threadIdx.x & 31;
  int tile = blockIdx.x * 4 + wave;
  if (tile < mtiles) {  // wave-uniform branch: EXEC stays all-ones for WMMA
    int row0 = tile * 16;
    int m = lane & 15;
    int kb = (lane < 16) ? 0 : 2;
    const float* hrow = H + (size_t)(row0 + m) * DD;

    v8f acc[4] = {v8f{}, v8f{}, v8f{}, v8f{}};
    for (int kk = 0; kk < 16; ++kk) {
      int k0 = kk * 4 + kb;
      v2f a;
      a.x = hrow[k0];
      a.y = hrow[k0 + 1];
#pragma unroll
      for (int j = 0; j < 4; ++j) {
        int n = j * 16 + m;
        v2f b;
        b.x = sW[k0 * DD + n];
        b.y = sW[(k0 + 1) * DD + n];
        acc[j] = __builtin_amdgcn_wmma_f32_16x16x4_f32(
            /*neg_a=*/false, a, /*neg_b=*/false, b,
            /*c_mod=*/(short)0, acc[j], /*reuse_a=*/false, /*reuse_b=*/false);
      }
    }
    int mbase = (lane < 16) ? 0 : 8;
#pragma unroll
    for (int j = 0; j < 4; ++j) {
#pragma unroll
      for (int r = 0; r < 8; ++r) {
        XW[(size_t)(row0 + mbase + r) * DD + j * 16 + m] = acc[j][r];
      }
    }
  }
}

// ---------------------------------------------------------------------------
// Fused per-node softmax + aggregation (gather form, zero output atomics).
// One 16-lane group per (t, n); lane q owns features [4q, 4q+3].
//   pass 1: m   = max_e  logit(t,e)
//   pass 2: s   = sum_e  exp(logit - m)
//   pass 3: acc = sum_e (exp(logit-m)/s) * sig_ef[etype,:] * xw[t,col,:]
// Empty nodes write zero (matches segment_sum), no division executed.
// xw_stride_t == 0 -> layer-0 shared XW (h0[t]==x for every t).
// ---------------------------------------------------------------------------
__global__ __launch_bounds__(256) void k_node_aggregate(
    const int* __restrict__ rowptr, const int* __restrict__ eord,
    const int* __restrict__ col, const int* __restrict__ et,
    const float* __restrict__ rgtab, const float* __restrict__ ri,
    const float* __restrict__ rj, const float* __restrict__ sig_ef,
    const float* __restrict__ xw, size_t xw_stride_t, float* __restrict__ out,
    int Nn) {
  int group = (int)(((long long)blockIdx.x * blockDim.x + threadIdx.x) >> 4);
  int q = threadIdx.x & 15;
  if (group >= TT * Nn) return;
  int t = group / Nn;
  int n = group - t * Nn;

  int beg = rowptr[n];
  int endp = rowptr[n + 1];
  float rin = ri[(size_t)t * Nn + n];
  const float* rjt = rj + (size_t)t * Nn;
  const float* rgt = rgtab + t * TT;

  // pass 1: segment max
  float m = -INFINITY;
  for (int i = beg; i < endp; ++i) {
    int e = eord[i];
    float lg = rgt[et[e]] + rin + rjt[col[e]];
    m = fmaxf(m, lg);
  }
  // pass 2: segment sum of exp
  float s = 0.0f;
  for (int i = beg; i < endp; ++i) {
    int e = eord[i];
    float lg = rgt[et[e]] + rin + rjt[col[e]];
    s += expf(lg - m);
  }
  // pass 3: weighted feature aggregation
  float4 acc = {0.0f, 0.0f, 0.0f, 0.0f};
  const float* xwt = xw + (size_t)t * xw_stride_t;
  for (int i = beg; i < endp; ++i) {
    int e = eord[i];
    int c = col[e], ty = et[e];
    float lg = rgt[ty] + rin + rjt[c];
    float w = expf(lg - m) / s;
    float4 sg = *(const float4*)(sig_ef + (size_t)ty * DD + q * 4);
    float4 xv = *(const float4*)(xwt + (size_t)c * DD + q * 4);
    acc.x += w * sg.x * xv.x;
    acc.y += w * sg.y * xv.y;
    acc.z += w * sg.z * xv.z;
    acc.w += w * sg.w * xv.w;
  }
  *(float4*)(out + ((size_t)t * Nn + n) * DD + q * 4) = acc;
}

// ---------------------------------------------------------------------------
// Residual update after layer 0 (in place): out = x(broadcast) + relu(out)
// ---------------------------------------------------------------------------
__global__ void k_residual_relu(const float* __restrict__ x,
                                float* __restrict__ out, size_t total,
                                size_t nd) {
  size_t g = (size_t)blockIdx.x * blockDim.x + threadIdx.x;
  if (g >= total) return;
  size_t t = g / nd;
  float o = out[g];
  out[g] = x[g - t * nd] + fmaxf(o, 0.0f);
}

// ---------------------------------------------------------------------------
// Host-side orchestration
// ---------------------------------------------------------------------------
extern "C" void kernel_launch(void* const* d_in, const int* in_sizes, int n_in,
                              void* d_out, int out_size, void* d_ws,
                              size_t ws_size, hipStream_t stream) {
  const float* x    = (const float*)d_in[0];
  const int*   eidx = (const int*)d_in[1];
  const int*   etyp = (const int*)d_in[2];
  const float* ef   = (const float*)d_in[3];
  const float* tg   = (const float*)d_in[4];
  const float* thi  = (const float*)d_in[5];
  const float* thj  = (const float*)d_in[6];
  const float* we   = (const float*)d_in[7];
  const float* wr   = (const float*)d_in[8];

  const int N = in_sizes[0] / DD;  // 50000
  const int E = in_sizes[2];       // 800000
  const int* row = eidx;           // edge_index[0] (destinations)
  const int* col = eidx + E;       // edge_index[1] (sources)

  // Workspace carve-out
  float* ws = (float*)d_ws;
  size_t off = 0;
  float* xw    = ws + off; off += (size_t)TT * N * DD;  // GEMM results
  float* hbuf  = ws + off; off += (size_t)TT * N * DD;  // out0 / h1
  float* ri    = ws + off; off += (size_t)TT * N;
  float* rj    = ws + off; off += (size_t)TT * N;
  float* efn0  = ws + off; off += TT * DD;
  float* efn1  = ws + off; off += TT * DD;
  float* sigef = ws + off; off += TT * DD;
  float* rgtab = ws + off; off += TT * TT;
  int* deg     = (int*)(ws + off); off += (size_t)N;
  int* cursor  = (int*)(ws + off); off += (size_t)N;
  int* rowptr  = (int*)(ws + off); off += (size_t)N + 1;
  int* eord    = (int*)(ws + off); off += (size_t)E;
  (void)ws_size; (void)n_in; (void)out_size;

  const size_t ND  = (size_t)N * DD;
  const size_t TND = (size_t)TT * ND;
  const int TB = 256;
  auto cdiv = [](long long a, long long b) { return (int)((a + b - 1) / b); };

  dim3 gridNode(cdiv(N, TB), TT);
  int gridE   = cdiv(E, TB);
  int gridN   = cdiv(N, TB);
  int gridTND = cdiv((long long)TND, TB);
  int gridAgg = cdiv((long long)TT * N * 16, TB);

  // =================== CSR build (shared by both layers) ===================
  k_fill_i32<<<gridN, TB, 0, stream>>>(deg, 0, (size_t)N);
  k_fill_i32<<<gridN, TB, 0, stream>>>(cursor, 0, (size_t)N);
  k_count_deg<<<gridE, TB, 0, stream>>>(row, deg, E);
  k_scan_rowptr<<<1, SCAN_THREADS, 0, stream>>>(deg, rowptr, N);
  k_bucket_edges<<<gridE, TB, 0, stream>>>(row, rowptr, cursor, eord, E);

  // ===================== Layer 0 (h0[t] = x broadcast) =====================
  k_small_params<<<1, 256, 0, stream>>>(ef, 0, tg, wr, efn0, sigef, rgtab);
  k_node_proj<<<gridNode, TB, 0, stream>>>(x, (size_t)0, thi, thj, ri, rj, N);
  {
    int mtiles = N / 16;  // xw shared across t in layer 0
    k_gemm_we<<<cdiv(mtiles, 4), 128, 0, stream>>>(x, we, xw, mtiles);
  }
  k_node_aggregate<<<gridAgg, TB, 0, stream>>>(rowptr, eord, col, etyp, rgtab,
                                               ri, rj, sigef, xw, (size_t)0,
                                               hbuf, N);
  k_residual_relu<<<gridTND, TB, 0, stream>>>(x, hbuf, TND, ND);  // hbuf = h1

  // ===================== Layer 1 (final: h = out) ==========================
  const float* tg1  = tg  + TT * DD;
  const float* thi1 = thi + TT * DD;
  const float* thj1 = thj + TT * DD;
  const float* we1  = we  + DD * DD;
  const float* wr1  = wr  + DD * DD;
  float* outp = (float*)d_out;

  k_small_params<<<1, 256, 0, stream>>>(efn0, 1, tg1, wr1, efn1, sigef, rgtab);
  k_node_proj<<<gridNode, TB, 0, stream>>>(hbuf, ND, thi1, thj1, ri, rj, N);
  {
    int mtiles = (TT * N) / 16;
    k_gemm_we<<<cdiv(mtiles, 4), 128, 0, stream>>>(hbuf, we1, xw, mtiles);
  }
  k_node_aggregate<<<gridAgg, TB, 0, stream>>>(rowptr, eord, col, etyp, rgtab,
                                               ri, rj, sigef, xw, ND, outp, N);
}